// ResidualVQ_12678743458280
// MI455X (gfx1250) — compile-verified
//
#include <hip/hip_runtime.h>
#include <hip/hip_bf16.h>
#include <math.h>

// Problem constants (from reference): x[B=32, C=512, T=2048], codebooks[Q=6, K=512, D=512]
#define BDIM 32
#define CDIM 512
#define TDIM 2048
#define QDIM 6
#define KDIM 512
#define DDIM 512
#define NVEC (BDIM * TDIM)                  // 65536 residual vectors per stage
#define NELEM ((size_t)BDIM * CDIM * TDIM)  // 33554432 elements
#define APAD 516                            // A-tile LDS row stride (bank-conflict pad, keeps 8B align)

typedef __attribute__((ext_vector_type(2))) float v2f;
typedef __attribute__((ext_vector_type(8))) float v8f;
typedef unsigned long long u64;

// Pack (distance, code index) into one sortable u64: order-preserving float->u32 map in the
// high word, index in the low word. min(u64) == (min distance, lowest index on ties).
__device__ __forceinline__ u64 pack_key(float v, int ci) {
  unsigned u = __float_as_uint(v);
  u = (u & 0x80000000u) ? ~u : (u | 0x80000000u);
  return ((u64)u << 32) | (unsigned)ci;
}

// xor-exchange via ds_swizzle group-of-32 mode: offset = {xor[14:10], or[9:5], and[4:0]}.
// xor masks < 16 stay inside each 16-lane half of the wave (matches WMMA C M-split).
// Branchless 64-bit min: v_cmp_lt_u64 + cndmask, no EXEC manipulation.
template <int XORMASK>
__device__ __forceinline__ u64 swz_min16(u64 k) {
  constexpr int imm = (XORMASK << 10) | 0x1f;
  unsigned olo = (unsigned)__builtin_amdgcn_ds_swizzle((int)(unsigned)k, imm);
  unsigned ohi = (unsigned)__builtin_amdgcn_ds_swizzle((int)(unsigned)(k >> 32), imm);
  u64 ok = ((u64)ohi << 32) | olo;
  return ok < k ? ok : k;
}

__device__ __forceinline__ u64 reduce_min16(u64 k) {
  k = swz_min16<8>(k);
  k = swz_min16<4>(k);
  k = swz_min16<2>(k);
  k = swz_min16<1>(k);
  return k;
}

// ---------------- residual init: R = x (R aliases d_out quantized region) ---------------
__global__ __launch_bounds__(256) void copy_res_kernel(const float4* __restrict__ x,
                                                       float4* __restrict__ r) {
  size_t i = (size_t)blockIdx.x * 256 + threadIdx.x;
  r[i] = x[i];
}

__global__ void zero_small_kernel(float* counts, float* lsum) {
  int t = threadIdx.x;
  for (int i = t; i < QDIM * KDIM; i += 256) counts[i] = 0.0f;
  if (t < QDIM) lsum[t] = 0.0f;
}

// ---------------- codebook pre-swizzle: cbP[q][(d>>1)*1024 + k*2 + (d&1)] = cb[q][k][d] ---
// Makes each WMMA B-fragment a single coalesced b64 load:
//   (B[k][code], B[k+1][code]) contiguous, consecutive lanes = consecutive codes.
__global__ __launch_bounds__(256) void prep_cb_kernel(const float* __restrict__ cb,
                                                      float* __restrict__ cbP) {
  size_t g = (size_t)blockIdx.x * 256 + threadIdx.x;
  int q = (int)(g >> 18);        // / (512*512)
  int r = (int)(g & 262143);
  int k = r >> 9;
  int d = r & 511;
  float v = cb[g];
  cbP[(size_t)q * 262144 + (size_t)(d >> 1) * 1024 + k * 2 + (d & 1)] = v;
}

// ---------------- per-code squared norms --------------------------------------------------
__global__ __launch_bounds__(128) void norms_kernel(const float* __restrict__ cb,
                                                    float* __restrict__ norms) {
  __shared__ float red[128];
  const float* row = cb + (size_t)blockIdx.x * DDIM;
  float s = 0.0f;
  for (int d = threadIdx.x; d < DDIM; d += 128) { float v = row[d]; s += v * v; }
  red[threadIdx.x] = s;
  __syncthreads();
  for (int off = 64; off > 0; off >>= 1) {
    if (threadIdx.x < off) red[threadIdx.x] += red[threadIdx.x + off];
    __syncthreads();
  }
  if (threadIdx.x == 0) norms[blockIdx.x] = red[0];
}

// ---------------- fused: distance GEMM (WMMA f32) + argmin + residual update + loss -------
// One block = 16 residual rows x all 512 codes. 8 waves, wave w owns codes [64w, 64w+64)
// as four 16x16 WMMA accumulator tiles. dist(n,k) = ||cb_k||^2 - 2*<x_n, cb_k>.
// Epilogue reuses the LDS-resident A tile to write new_r = r - cb[argmin] back (one read +
// one write of the residual per stage total), accumulating commit loss = sum(new_r^2).
__global__ __launch_bounds__(256)
void vq_fused_kernel(float* __restrict__ R, const float* __restrict__ cbP,
                     const float* __restrict__ cbq, const float* __restrict__ norms,
                     float* __restrict__ idx_out, float* __restrict__ counts,
                     float* __restrict__ lsum) {
  __shared__ float Alds[16 * APAD];  // residual tile [row][d], survives until epilogue
  __shared__ u64 pkey[16 * 32];      // per-(row, wave*4+ct) partial min keys
  __shared__ int bsel[16];           // final argmin per row
  __shared__ float redsum[256];      // loss reduction

  const int tid = threadIdx.x;
  const int lane = tid & 31;
  const int w = tid >> 5;
  const int n0 = blockIdx.x * 16;
  const int b = n0 >> 11;            // n0 / T
  const int t0 = n0 & (TDIM - 1);
  const int r16 = tid & 15;          // row for staging/epilogue (lanes fast -> t coalesced)
  const int dg = tid >> 4;           // dim group 0..15

  const float* gbase = R + (size_t)b * CDIM * TDIM + t0 + r16;

  // Stage A: Alds[r][d] = R[b, d, t0+r]; consecutive lanes -> consecutive t (coalesced)
  for (int j = 0; j < 32; ++j) {
    int d = j * 16 + dg;
    Alds[r16 * APAD + d] = gbase[(size_t)d * TDIM];
  }
  __syncthreads();

  // 32-bit A 16x4 lane layout: lanes 0-15 hold M=lane with K={k0,k0+1};
  // lanes 16-31 hold M=lane-16 with K={k0+2,k0+3}.
  const int row = lane & 15;
  const int khalf = (lane >> 4) << 1;
  const int c0 = w * 64;

  v8f acc0 = {}, acc1 = {}, acc2 = {}, acc3 = {};
  const float* bbase = cbP + (size_t)(khalf >> 1) * 1024 + (size_t)(c0 + row) * 2;

#pragma unroll 4
  for (int k0 = 0; k0 < DDIM; k0 += 4) {
    v2f a = *(const v2f*)&Alds[row * APAD + k0 + khalf];
    const float* bk = bbase + (size_t)(k0 >> 1) * 1024;
    v2f b0 = *(const v2f*)(bk);
    v2f b1 = *(const v2f*)(bk + 32);
    v2f b2 = *(const v2f*)(bk + 64);
    v2f b3 = *(const v2f*)(bk + 96);
    acc0 = __builtin_amdgcn_wmma_f32_16x16x4_f32(false, a, false, b0, (short)0, acc0, false, false);
    acc1 = __builtin_amdgcn_wmma_f32_16x16x4_f32(false, a, false, b1, (short)0, acc1, false, false);
    acc2 = __builtin_amdgcn_wmma_f32_16x16x4_f32(false, a, false, b2, (short)0, acc2, false, false);
    acc3 = __builtin_amdgcn_wmma_f32_16x16x4_f32(false, a, false, b3, (short)0, acc3, false, false);
  }

  // In-register argmin across each tile's 16 columns via ds_swizzle xor-exchanges on packed keys.
  // C/D layout: lanes 0-15 col=lane, VGPR j -> M=j ; lanes 16-31 col=lane-16, M=j+8.
  {
    const int col = lane & 15;
    const int mhi = (lane >> 4) << 3;
#define VQ_ARGMIN_TILE(ACC, CT)                                              \
    {                                                                        \
      int code = c0 + (CT)*16 + col;                                         \
      float nrm = norms[code];                                               \
      _Pragma("unroll") for (int j = 0; j < 8; ++j) {                        \
        u64 key = pack_key(nrm - 2.0f * (ACC)[j], code);                     \
        key = reduce_min16(key);                                             \
        if (col == 0) pkey[(mhi + j) * 32 + w * 4 + (CT)] = key;             \
      }                                                                      \
    }
    VQ_ARGMIN_TILE(acc0, 0) VQ_ARGMIN_TILE(acc1, 1)
    VQ_ARGMIN_TILE(acc2, 2) VQ_ARGMIN_TILE(acc3, 3)
#undef VQ_ARGMIN_TILE
  }
  __syncthreads();

  // Final reduce: 16 groups of 16 lanes; group g reduces row g's 32 partial keys.
  {
    const int grow = tid >> 4;
    const int gl = tid & 15;
    u64 k1 = pkey[grow * 32 + gl];
    u64 k2 = pkey[grow * 32 + 16 + gl];
    u64 key = k2 < k1 ? k2 : k1;
    key = reduce_min16(key);
    int ci = (int)(unsigned)key;
    if (gl == 0) {
      idx_out[n0 + grow] = (float)ci;
      atomicAdd(&counts[ci], 1.0f);
      bsel[grow] = ci;
    }
  }
  __syncthreads();

  // Epilogue: new_r = A - cb[idx]; write back (t-coalesced), accumulate loss = sum(new_r^2).
  // The 16 gathered codebook rows (32KB) are served from WGP$/L2.
  {
    const float* crow = cbq + (size_t)bsel[r16] * DDIM;
    float lacc = 0.0f;
    float* wbase = R + (size_t)b * CDIM * TDIM + t0 + r16;
    for (int j = 0; j < 32; ++j) {
      int d = j * 16 + dg;
      float nr = Alds[r16 * APAD + d] - crow[d];
      wbase[(size_t)d * TDIM] = nr;
      lacc += nr * nr;
    }
    redsum[tid] = lacc;
  }
  __syncthreads();
  for (int off = 128; off > 0; off >>= 1) {
    if (tid < off) redsum[tid] += redsum[tid + off];
    __syncthreads();
  }
  if (tid == 0) atomicAdd(lsum, redsum[0]);
}

// ---------------- quantized_out = x - residual_final (in place over d_out) ----------------
__global__ __launch_bounds__(256) void qout_kernel(const float4* __restrict__ x,
                                                   float4* __restrict__ o) {
  size_t i = (size_t)blockIdx.x * 256 + threadIdx.x;
  float4 a = x[i];
  float4 r = o[i];
  float4 q;
  q.x = a.x - r.x; q.y = a.y - r.y; q.z = a.z - r.z; q.w = a.w - r.w;
  o[i] = q;
}

// ---------------- scalars: mean commit loss, mean perplexity ------------------------------
__global__ void finalize_kernel(const float* __restrict__ counts, const float* __restrict__ lsum,
                                float* __restrict__ scal) {
  if (blockIdx.x == 0 && threadIdx.x == 0) {
    float lm = 0.0f;
    for (int q = 0; q < QDIM; ++q) lm += lsum[q] * (1.0f / (float)NELEM);
    lm *= (1.0f / QDIM);
    float pm = 0.0f;
    for (int q = 0; q < QDIM; ++q) {
      float ent = 0.0f;
      for (int k = 0; k < KDIM; ++k) {
        float p = counts[q * KDIM + k] * (1.0f / (float)NVEC);
        ent -= p * logf(p + 1e-10f);
      }
      pm += expf(ent);
    }
    pm *= (1.0f / QDIM);
    scal[0] = lm;
    scal[1] = pm;
  }
}

extern "C" void kernel_launch(void* const* d_in, const int* in_sizes, int n_in,
                              void* d_out, int out_size, void* d_ws, size_t ws_size,
                              hipStream_t stream) {
  const float* x = (const float*)d_in[0];       // [32, 512, 2048]
  const float* cb = (const float*)d_in[1];      // [6, 512, 512]
  float* out = (float*)d_out;
  float* Rbuf = out;                            // residual lives in the quantized region
  float* idx_out = out + NELEM;                 // [6, 65536] indices (as float)
  float* scal = idx_out + (size_t)QDIM * NVEC;  // [loss, perplexity]

  float* cbP = (float*)d_ws;                        // 6*512*512 swizzled codebook
  float* norms = cbP + (size_t)QDIM * KDIM * DDIM;  // 6*512
  float* counts = norms + QDIM * KDIM;              // 6*512
  float* lsum = counts + QDIM * KDIM;               // 6

  copy_res_kernel<<<(int)(NELEM / 4 / 256), 256, 0, stream>>>((const float4*)x, (float4*)Rbuf);
  zero_small_kernel<<<1, 256, 0, stream>>>(counts, lsum);
  prep_cb_kernel<<<QDIM * KDIM * DDIM / 256, 256, 0, stream>>>(cb, cbP);
  norms_kernel<<<QDIM * KDIM, 128, 0, stream>>>(cb, norms);

  for (int q = 0; q < QDIM; ++q) {
    vq_fused_kernel<<<NVEC / 16, 256, 0, stream>>>(
        Rbuf, cbP + (size_t)q * KDIM * DDIM, cb + (size_t)q * KDIM * DDIM,
        norms + q * KDIM, idx_out + (size_t)q * NVEC, counts + q * KDIM, lsum + q);
  }

  qout_kernel<<<(int)(NELEM / 4 / 256), 256, 0, stream>>>((const float4*)x, (float4*)out);
  finalize_kernel<<<1, 1, 0, stream>>>(counts, lsum, scal);
}